// EncodecEuclideanCodebook_23356032156247
// MI455X (gfx1250) — compile-verified
//
#include <hip/hip_runtime.h>

typedef float v2f __attribute__((ext_vector_type(2)));
typedef float v4f __attribute__((ext_vector_type(4)));
typedef float v8f __attribute__((ext_vector_type(8)));
typedef int   b128_t __attribute__((vector_size(16)));   // matches builtin's V4i

#define DDIM 128            // embedding dim
#define KCB  2048           // codebook size
#define NROWS (32 * 4096)   // B*T
#define KC   128            // codebook rows per LDS chunk
#define NCHUNK (KCB / KC)   // 16
#define LDS_STRIDE 132      // padded row stride (floats) -> bank-conflict-free b64 loads
#define BUF_FLOATS (KC * LDS_STRIDE)
#define MBLOCK 256          // rows per workgroup (8 waves x 32 rows)

// ---- CDNA5 async global->LDS copy (ASYNCcnt), 16B per lane ----
__device__ __forceinline__ void async_copy_b128(const float* g, float* l) {
#if __has_builtin(__builtin_amdgcn_global_load_async_to_lds_b128)
    __builtin_amdgcn_global_load_async_to_lds_b128(
        (__attribute__((address_space(1))) b128_t*)g,
        (__attribute__((address_space(3))) b128_t*)l,
        0, 0);
#else
    unsigned ldsoff = (unsigned)(unsigned long long)
        (__attribute__((address_space(3))) float*)l;
    asm volatile("global_load_async_to_lds_b128 %0, %1, off"
                 :: "v"(ldsoff), "v"(g) : "memory");
#endif
}

__device__ __forceinline__ void wait_async0() {
#if __has_builtin(__builtin_amdgcn_s_wait_asynccnt)
    __builtin_amdgcn_s_wait_asynccnt(0);
#else
    asm volatile("s_wait_asynccnt 0" ::: "memory");
#endif
}

// stage one 128x128 f32 codebook chunk into a padded LDS buffer (async)
__device__ __forceinline__ void stage_chunk_async(const float* __restrict__ src,
                                                  float* dst, int tid) {
    int r0  = tid >> 5;          // 0..7
    int seg = tid & 31;          // 0..31 (16B segments per row)
    const float* g = src + (size_t)r0 * DDIM + seg * 4;
    float*       l = dst + r0 * LDS_STRIDE + seg * 4;
#pragma unroll
    for (int i = 0; i < 16; ++i)
        async_copy_b128(g + i * 8 * DDIM, l + i * 8 * LDS_STRIDE);
}

// --- kernel 1: h2[k] = 0.5 * ||embed[k]||^2  (one wave per codebook row) ---
__global__ void __launch_bounds__(256)
vq_e2_kernel(const float* __restrict__ embed, float* __restrict__ h2) {
    int wave = threadIdx.x >> 5;
    int lane = threadIdx.x & 31;
    int row  = blockIdx.x * 8 + wave;
    v4f v = *reinterpret_cast<const v4f*>(embed + (size_t)row * DDIM + lane * 4);
    float s = v.x * v.x + v.y * v.y + v.z * v.z + v.w * v.w;
#pragma unroll
    for (int m = 16; m >= 1; m >>= 1) s += __shfl_xor(s, m, 32);
    if (lane == 0) h2[row] = 0.5f * s;
}

// --- kernel 2: WMMA f32 GEMM + running argmax -> out_idx (as float) ---
// Double-buffered LDS chunks staged with async global->LDS copies; each wave
// owns TWO 16-row A tiles so every LDS B-pair feeds 2 WMMAs.
__global__ void __launch_bounds__(256)
vq_argmin_kernel(const float* __restrict__ hs, const float* __restrict__ embed,
                 const float* __restrict__ h2, float* __restrict__ out_idx) {
    extern __shared__ float smem[];
    float* buf0 = smem;
    float* buf1 = smem + BUF_FLOATS;

    const int tid  = threadIdx.x;
    const int wave = tid >> 5;
    const int lane = tid & 31;
    const int l15  = lane & 15;
    const int hi   = lane >> 4;              // 0: k-offs {0,1}, 1: k-offs {2,3}
    const int row_base = blockIdx.x * MBLOCK + wave * 32;

    // fire chunk-0 staging first so it overlaps the A-tile loads
    stage_chunk_async(embed, buf0, tid);

    // Two A tiles (rows [0,16) and [16,32)) in V_WMMA_F32_16X16X4_F32 A layout.
    // hs is a single-pass 64MB stream -> non-temporal loads.
    v2f a0[32], a1[32];
    const float* ap0 = hs + (size_t)(row_base + l15) * DDIM + (hi << 1);
    const float* ap1 = ap0 + 16 * DDIM;
#pragma unroll
    for (int kc = 0; kc < 32; ++kc) {
        a0[kc] = __builtin_nontemporal_load(reinterpret_cast<const v2f*>(ap0 + kc * 4));
        a1[kc] = __builtin_nontemporal_load(reinterpret_cast<const v2f*>(ap1 + kc * 4));
    }

    float best_v0[8], best_v1[8];
    int   best_i0[8], best_i1[8];
#pragma unroll
    for (int j = 0; j < 8; ++j) {
        best_v0[j] = -3.402823466e38f; best_i0[j] = 0;
        best_v1[j] = -3.402823466e38f; best_i1[j] = 0;
    }

    wait_async0();
    __syncthreads();

#pragma unroll 1
    for (int c = 0; c < NCHUNK; ++c) {
        float* cur = (c & 1) ? buf1 : buf0;
        if (c + 1 < NCHUNK)   // prefetch next chunk into the other buffer
            stage_chunk_async(embed + (size_t)(c + 1) * KC * DDIM,
                              (c & 1) ? buf0 : buf1, tid);

#pragma unroll 1
        for (int nt = 0; nt < KC / 16; ++nt) {
            const float* bp = cur + (nt * 16 + l15) * LDS_STRIDE + (hi << 1);
            v8f acc0 = {0.f, 0.f, 0.f, 0.f, 0.f, 0.f, 0.f, 0.f};
            v8f acc1 = {0.f, 0.f, 0.f, 0.f, 0.f, 0.f, 0.f, 0.f};
#pragma unroll
            for (int kc = 0; kc < 32; ++kc) {
                v2f b = *reinterpret_cast<const v2f*>(bp + kc * 4);
                acc0 = __builtin_amdgcn_wmma_f32_16x16x4_f32(
                        false, a0[kc], false, b, (short)0, acc0, false, false);
                acc1 = __builtin_amdgcn_wmma_f32_16x16x4_f32(
                        false, a1[kc], false, b, (short)0, acc1, false, false);
            }
            int   col = c * KC + nt * 16 + l15;
            float h2v = h2[col];             // 8KB L2-hot bias array
#pragma unroll
            for (int j = 0; j < 8; ++j) {
                float s0 = acc0[j] - h2v;    // dot - 0.5*||e||^2 (order == reference)
                float s1 = acc1[j] - h2v;
                if (s0 > best_v0[j]) { best_v0[j] = s0; best_i0[j] = col; }
                if (s1 > best_v1[j]) { best_v1[j] = s1; best_i1[j] = col; }
            }
        }

        wait_async0();     // my async stores into the other buffer are done
        __syncthreads();   // everyone finished reading `cur` + sees staged data
    }

    // argmax across the 16 lanes sharing each row; lowest index wins ties
#pragma unroll
    for (int j = 0; j < 8; ++j) {
        float v0 = best_v0[j]; int i0 = best_i0[j];
        float v1 = best_v1[j]; int i1 = best_i1[j];
#pragma unroll
        for (int m = 8; m >= 1; m >>= 1) {
            float ov0 = __shfl_xor(v0, m, 16);
            int   oi0 = __shfl_xor(i0, m, 16);
            if (ov0 > v0 || (ov0 == v0 && oi0 < i0)) { v0 = ov0; i0 = oi0; }
            float ov1 = __shfl_xor(v1, m, 16);
            int   oi1 = __shfl_xor(i1, m, 16);
            if (ov1 > v1 || (ov1 == v1 && oi1 < i1)) { v1 = ov1; i1 = oi1; }
        }
        if (l15 == 0) {
            __builtin_nontemporal_store((float)i0, &out_idx[row_base + j + 8 * hi]);
            __builtin_nontemporal_store((float)i1, &out_idx[row_base + 16 + j + 8 * hi]);
        }
    }
}

// --- kernel 3: decode gather, one wave per row (codebook stays hot in L2) ---
__global__ void __launch_bounds__(256)
vq_gather_kernel(const float* __restrict__ embed, const float* __restrict__ idxf,
                 float* __restrict__ quant) {
    int wave = threadIdx.x >> 5;
    int lane = threadIdx.x & 31;
    int row  = blockIdx.x * 8 + wave;
    int k    = (int)idxf[row];
    v4f v = *(reinterpret_cast<const v4f*>(embed + (size_t)k * DDIM) + lane);
    __builtin_nontemporal_store(
        v, reinterpret_cast<v4f*>(quant + (size_t)row * DDIM) + lane);
}

extern "C" void kernel_launch(void* const* d_in, const int* in_sizes, int n_in,
                              void* d_out, int out_size, void* d_ws, size_t ws_size,
                              hipStream_t stream) {
    const float* hs    = (const float*)d_in[0];   // [B,T,D] f32
    const float* embed = (const float*)d_in[1];   // [K,D]  f32
    float* out = (float*)d_out;                   // [N] indices (as float) ++ [N,D] quantize
    float* h2  = (float*)d_ws;                    // K floats scratch

    size_t shmem = (size_t)(2 * BUF_FLOATS) * sizeof(float);
    // allow >64KB dynamic LDS (idempotent; not a stream op, capture-safe)
    (void)hipFuncSetAttribute((const void*)vq_argmin_kernel,
                              hipFuncAttributeMaxDynamicSharedMemorySize,
                              (int)shmem);

    vq_e2_kernel<<<KCB / 8, 256, 0, stream>>>(embed, h2);
    vq_argmin_kernel<<<NROWS / MBLOCK, 256, shmem, stream>>>(hs, embed, h2, out);
    vq_gather_kernel<<<NROWS / 8, 256, 0, stream>>>(embed, out, out + NROWS);
}